// static_heto_graph_41558103556308
// MI455X (gfx1250) — compile-verified
//
#include <hip/hip_runtime.h>

typedef float v2f __attribute__((ext_vector_type(2)));
typedef float v8f __attribute__((ext_vector_type(8)));

// ---------------- workspace zeroing ----------------
__global__ void zero_f32(float* __restrict__ p, long long n) {
  long long i = (long long)blockIdx.x * blockDim.x + threadIdx.x;
  if (i < n) p[i] = 0.0f;
}

// ---------------- degree counting (both endpoints in one pass) ----------------
__global__ void deg_count(const int* __restrict__ src, const int* __restrict__ dst,
                          float* __restrict__ degS, float* __restrict__ degD, int nE) {
  int i = blockIdx.x * blockDim.x + threadIdx.x;
  if (i < nE) {
    atomicAdd(&degS[src[i]], 1.0f);
    atomicAdd(&degD[dst[i]], 1.0f);
  }
}

// deg -> clamp(deg,1)^-1/2, in place over the whole contiguous norm region
__global__ void deg_to_norm(float* __restrict__ p, int n) {
  int i = blockIdx.x * blockDim.x + threadIdx.x;
  if (i < n) p[i] = rsqrtf(fmaxf(p[i], 1.0f));
}

// ---------------- edge gather/scatter: agg[dst] += h[src] * norm_src[src] ----------------
__global__ void edge_scatter(const float* __restrict__ h,
                             const int* __restrict__ src, const int* __restrict__ dst,
                             const float* __restrict__ normS,
                             float* __restrict__ agg, int nE) {
  int i = blockIdx.x * blockDim.x + threadIdx.x;
  if (i >= nE) return;
  int s = src[i];
  int d = dst[i];
  float ns = normS[s];
  const float* x = h + (size_t)s * 5;
  float* a = agg + (size_t)d * 5;
  atomicAdd(a + 0, x[0] * ns);
  atomicAdd(a + 1, x[1] * ns);
  atomicAdd(a + 2, x[2] * ns);
  atomicAdd(a + 3, x[3] * ns);
  atomicAdd(a + 4, x[4] * ns);
}

// ---------------- finalize: out = (agg1*norm1)@W1 + b1 [+ (agg2*norm2)@W2 + b2] ----------------
// One wave handles a 16-node tile. d=5 is done as K=4 WMMA + K=1 WMMA (exact fp32).
// A layout (16x4 f32): lanes 0-15 -> K={0,1} in {a.x,a.y}; lanes 16-31 -> K={2,3}.
// B layout (4x16 f32): VGPR0 = rows K=0 (lanes 0-15) / K=2 (lanes 16-31); VGPR1 = K=1/K=3.
// C/D layout (16x16 f32): VGPR v -> M=v (lanes 0-15, N=lane) and M=v+8 (lanes 16-31, N=lane-16).
__global__ void finalize_wmma(const float* __restrict__ agg1, const float* __restrict__ norm1,
                              const float* __restrict__ W1, const float* __restrict__ b1,
                              const float* __restrict__ agg2, const float* __restrict__ norm2,
                              const float* __restrict__ W2, const float* __restrict__ b2,
                              float* __restrict__ out, int n, int two) {
  const int lane = threadIdx.x & 31;
  const int wave = (blockIdx.x * blockDim.x + threadIdx.x) >> 5;
  const int base = wave * 16;
  if (base >= n) return;            // wave-uniform: full waves proceed with EXEC all-ones

  const int idx = lane & 15;        // M index for A loads; N index for B/C/D
  const int hi  = lane >> 4;        // 0: K={0,1}(+K4); 1: K={2,3}
  const int node = base + idx;
  const bool valid = node < n;

  // C accumulator preloaded with bias (broadcast down the M rows)
  v8f c = {};
  float bias = 0.0f;
  if (idx < 5) {
    bias = b1[idx];
    if (two) bias += b2[idx];
  }
#pragma unroll
  for (int v = 0; v < 8; ++v) c[v] = bias;

  // ---- edge type 1 ----
  v2f aK01 = {0.0f, 0.0f}, aK4 = {0.0f, 0.0f};
  if (valid) {
    float nm = norm1[node];
    const float* x = agg1 + (size_t)node * 5;
    if (hi == 0) { aK01.x = x[0] * nm; aK01.y = x[1] * nm; aK4.x = x[4] * nm; }
    else         { aK01.x = x[2] * nm; aK01.y = x[3] * nm; }
  }
  v2f bK01 = {0.0f, 0.0f}, bK4 = {0.0f, 0.0f};
  if (idx < 5) {
    if (hi == 0) { bK01.x = W1[0 * 5 + idx]; bK01.y = W1[1 * 5 + idx]; bK4.x = W1[4 * 5 + idx]; }
    else         { bK01.x = W1[2 * 5 + idx]; bK01.y = W1[3 * 5 + idx]; }
  }
  c = __builtin_amdgcn_wmma_f32_16x16x4_f32(false, aK01, false, bK01, (short)0, c, false, false);
  c = __builtin_amdgcn_wmma_f32_16x16x4_f32(false, aK4,  false, bK4,  (short)0, c, false, false);

  // ---- edge type 2 (uniform branch) ----
  if (two) {
    v2f a2K01 = {0.0f, 0.0f}, a2K4 = {0.0f, 0.0f};
    if (valid) {
      float nm = norm2[node];
      const float* x = agg2 + (size_t)node * 5;
      if (hi == 0) { a2K01.x = x[0] * nm; a2K01.y = x[1] * nm; a2K4.x = x[4] * nm; }
      else         { a2K01.x = x[2] * nm; a2K01.y = x[3] * nm; }
    }
    v2f b2K01 = {0.0f, 0.0f}, b2K4 = {0.0f, 0.0f};
    if (idx < 5) {
      if (hi == 0) { b2K01.x = W2[0 * 5 + idx]; b2K01.y = W2[1 * 5 + idx]; b2K4.x = W2[4 * 5 + idx]; }
      else         { b2K01.x = W2[2 * 5 + idx]; b2K01.y = W2[3 * 5 + idx]; }
    }
    c = __builtin_amdgcn_wmma_f32_16x16x4_f32(false, a2K01, false, b2K01, (short)0, c, false, false);
    c = __builtin_amdgcn_wmma_f32_16x16x4_f32(false, a2K4,  false, b2K4,  (short)0, c, false, false);
  }

  // ---- store D: only N<5 columns are real ----
  if (idx < 5) {
#pragma unroll
    for (int v = 0; v < 8; ++v) {
      int m = base + v + hi * 8;
      if (m < n) out[(size_t)m * 5 + idx] = c[v];
    }
  }
}

static inline int cdiv(long long a, long long b) { return (int)((a + b - 1) / b); }

extern "C" void kernel_launch(void* const* d_in, const int* in_sizes, int n_in,
                              void* d_out, int out_size, void* d_ws, size_t ws_size,
                              hipStream_t stream) {
  // inputs (setup_inputs order)
  const float* h_word  = (const float*)d_in[1];
  const float* h_topic = (const float*)d_in[2];
  const float* W_wt = (const float*)d_in[3];
  const float* W_ww = (const float*)d_in[4];
  const float* W_wd = (const float*)d_in[5];
  const float* W_td = (const float*)d_in[6];
  const float* W_tt = (const float*)d_in[7];
  const float* b_wt = (const float*)d_in[8];
  const float* b_ww = (const float*)d_in[9];
  const float* b_wd = (const float*)d_in[10];
  const float* b_td = (const float*)d_in[11];
  const float* b_tt = (const float*)d_in[12];
  const int* ww_src = (const int*)d_in[13];
  const int* ww_dst = (const int*)d_in[14];
  const int* wt_src = (const int*)d_in[15];
  const int* wt_dst = (const int*)d_in[16];
  const int* wd_src = (const int*)d_in[17];
  const int* wd_dst = (const int*)d_in[18];
  const int* td_src = (const int*)d_in[19];
  const int* td_dst = (const int*)d_in[20];
  const int* tt_src = (const int*)d_in[21];
  const int* tt_dst = (const int*)d_in[22];

  const int ND = in_sizes[0] / 5;
  const int NW = in_sizes[1] / 5;
  const int NT = in_sizes[2] / 5;
  const int E_ww = in_sizes[13];
  const int E_wt = in_sizes[15];
  const int E_wd = in_sizes[17];
  const int E_td = in_sizes[19];
  const int E_tt = in_sizes[21];

  // ---- workspace layout (floats): 10 norm arrays (contiguous), then 5 agg arrays ----
  float* ws = (float*)d_ws;
  size_t o = 0;
  float* n_ww_s = ws + o; o += NW;
  float* n_ww_d = ws + o; o += NW;
  float* n_wt_s = ws + o; o += NW;
  float* n_wt_d = ws + o; o += NT;
  float* n_wd_s = ws + o; o += NW;
  float* n_wd_d = ws + o; o += ND;
  float* n_td_s = ws + o; o += NT;
  float* n_td_d = ws + o; o += ND;
  float* n_tt_s = ws + o; o += NT;
  float* n_tt_d = ws + o; o += NT;
  const size_t NORM_TOTAL = o;
  float* agg_ww = ws + o; o += (size_t)NW * 5;
  float* agg_wt = ws + o; o += (size_t)NT * 5;
  float* agg_wd = ws + o; o += (size_t)ND * 5;
  float* agg_td = ws + o; o += (size_t)ND * 5;
  float* agg_tt = ws + o; o += (size_t)NT * 5;
  const long long TOT = (long long)o;

  const int B = 256;

  // 1) zero degrees + accumulators
  zero_f32<<<cdiv(TOT, B), B, 0, stream>>>(ws, TOT);

  // 2) per-etype degree counting (src out-degree and dst in-degree in one pass)
  deg_count<<<cdiv(E_ww, B), B, 0, stream>>>(ww_src, ww_dst, n_ww_s, n_ww_d, E_ww);
  deg_count<<<cdiv(E_wt, B), B, 0, stream>>>(wt_src, wt_dst, n_wt_s, n_wt_d, E_wt);
  deg_count<<<cdiv(E_wd, B), B, 0, stream>>>(wd_src, wd_dst, n_wd_s, n_wd_d, E_wd);
  deg_count<<<cdiv(E_td, B), B, 0, stream>>>(td_src, td_dst, n_td_s, n_td_d, E_td);
  deg_count<<<cdiv(E_tt, B), B, 0, stream>>>(tt_src, tt_dst, n_tt_s, n_tt_d, E_tt);

  // 3) deg -> clamp(deg,1)^-1/2 over the whole contiguous norm region
  deg_to_norm<<<cdiv((long long)NORM_TOTAL, B), B, 0, stream>>>(ws, (int)NORM_TOTAL);

  // 4) edge gather/scatter with src normalization (L2-resident atomics)
  edge_scatter<<<cdiv(E_ww, B), B, 0, stream>>>(h_word,  ww_src, ww_dst, n_ww_s, agg_ww, E_ww);
  edge_scatter<<<cdiv(E_wt, B), B, 0, stream>>>(h_word,  wt_src, wt_dst, n_wt_s, agg_wt, E_wt);
  edge_scatter<<<cdiv(E_wd, B), B, 0, stream>>>(h_word,  wd_src, wd_dst, n_wd_s, agg_wd, E_wd);
  edge_scatter<<<cdiv(E_td, B), B, 0, stream>>>(h_topic, td_src, td_dst, n_td_s, agg_td, E_td);
  edge_scatter<<<cdiv(E_tt, B), B, 0, stream>>>(h_topic, tt_src, tt_dst, n_tt_s, agg_tt, E_tt);

  // 5) WMMA finalize: out tuple is (doc, word, topic) concatenated flat
  float* out_doc   = (float*)d_out;
  float* out_word  = out_doc  + (size_t)ND * 5;
  float* out_topic = out_word + (size_t)NW * 5;

  {
    long long th = (long long)((ND + 15) / 16) * 32;
    finalize_wmma<<<cdiv(th, B), B, 0, stream>>>(agg_wd, n_wd_d, W_wd, b_wd,
                                                 agg_td, n_td_d, W_td, b_td,
                                                 out_doc, ND, 1);
  }
  {
    long long th = (long long)((NW + 15) / 16) * 32;
    finalize_wmma<<<cdiv(th, B), B, 0, stream>>>(agg_ww, n_ww_d, W_ww, b_ww,
                                                 nullptr, nullptr, nullptr, nullptr,
                                                 out_word, NW, 0);
  }
  {
    long long th = (long long)((NT + 15) / 16) * 32;
    finalize_wmma<<<cdiv(th, B), B, 0, stream>>>(agg_wt, n_wt_d, W_wt, b_wt,
                                                 agg_tt, n_tt_d, W_tt, b_tt,
                                                 out_topic, NT, 1);
  }
}